// SelfmixLayer_68985764708564
// MI455X (gfx1250) — compile-verified
//
#include <hip/hip_runtime.h>

typedef __attribute__((ext_vector_type(2))) float v2f;
typedef __attribute__((ext_vector_type(8))) float v8f;
typedef __attribute__((ext_vector_type(4))) float f4;

// Problem constants (from reference): Lin=2, Lout=3, Fin=128, N=32768, Mi=5, Mo=7
#define N_  32768
#define F_  128
#define MI_ 5
#define MO_ 7

// Packed output columns c=0..15 -> (l,m): 0:(0,0); 1-3:(1,0..2); 4-8:(2,0..4); 9-15:(3,0..6)
__device__ __forceinline__ void col_to_lm(int c, int& l, int& m) {
    if (c == 0)      { l = 0; m = 0; }
    else if (c < 4)  { l = 1; m = c - 1; }
    else if (c < 9)  { l = 2; m = c - 4; }
    else             { l = 3; m = c - 9; }
}

// B matrix [32 x 16]: rows 0..22 = packed valid CG coefficients, rows 23..31 = identity
// block routing the 9 self-interaction products to columns 0..8.
__device__ __forceinline__ float load_b(int kk, int c, const float* __restrict__ cgc) {
    if (kk >= 23) return (kk - 23 == c) ? 1.0f : 0.0f;
    int l, m; col_to_lm(c, l, m);
    int l1, l2, p, q;
    if (kk < 3)      { l1 = 0; l2 = 1; p = 0;            q = kk;           }  // pair (0,1): 1x3
    else if (kk < 8) { l1 = 0; l2 = 2; p = 0;            q = kk - 3;       }  // pair (0,2): 1x5
    else             { l1 = 1; l2 = 2; p = (kk - 8) / 5; q = (kk - 8) % 5; }  // pair (1,2): 3x5
    // cgc dims [3][3][4][5][5][7]: cgc[l1][l2][l][p][q][m]
    int idx = ((((l1 * 3 + l2) * 4 + l) * 5 + p) * 5 + q) * 7 + m;
    return cgc[idx];
}

__global__ void __launch_bounds__(256)
selfmix_wmma_kernel(const float* __restrict__ x,   // [3, N, 5, 128]
                    const float* __restrict__ k,   // [4, 128]
                    const float* __restrict__ s,   // [12, 128]
                    const float* __restrict__ cgc, // [3,3,4,5,5,7]
                    float* __restrict__ out) {     // [4, N, 7, 128]
    const int lane = threadIdx.x & 31;
    const int gw   = (int)((blockIdx.x * blockDim.x + threadIdx.x) >> 5); // global wave id
    const int nW   = (int)((gridDim.x * blockDim.x) >> 5);                // total waves
    const int f0   = (gw & 7) << 4;    // this wave's fixed 16-channel slice
    const int n0   = gw >> 3;          // first point
    const int nStride = nW >> 3;       // points step
    const int c  = lane & 15;          // B/D column for this lane
    const int hi = lane >> 4;

    // ---- Loop-invariant: B operand (packed cgc + identity), gathered once ----
    v2f B[8];
#pragma unroll
    for (int t = 0; t < 8; ++t) {
        const int k0 = 4 * t + 2 * hi;        // lanes<16: K=4t,4t+1 ; lanes>=16: K=4t+2,4t+3
        B[t].x = load_b(k0,     c, cgc);
        B[t].y = load_b(k0 + 1, c, cgc);
    }

    // ---- Loop-invariant: per-lane s and k (depend only on channel f0+c) ----
    const int fA  = f0 + c;
    const float s01 = s[1 * F_ + fA];   // s[l1*Lin + l2], (0,1)->1
    const float s02 = s[2 * F_ + fA];   // (0,2)->2
    const float s12 = s[4 * F_ + fA];   // (1,2)->4
    const float kk0 = k[0 * F_ + fA];
    const float kk1 = k[1 * F_ + fA];
    const float kk2 = k[2 * F_ + fA];

    // ---- Loop-invariant: store-side decode ----
    int lD, mD; col_to_lm(c, lD, mD);
    int lZ = 0, mZ = 0;
    const int doZero = (lane < 24);
    if (doZero) {
        const int sl = lane >> 1;                    // invalid slot 0..11
        if (sl < 6)       { lZ = 0; mZ = sl + 1; }   // (0, 1..6)
        else if (sl < 10) { lZ = 1; mZ = sl - 3; }   // (1, 3..6)
        else              { lZ = 2; mZ = sl - 5; }   // (2, 5..6)
    }
    const int hf = lane & 1;
    const bool h = (hi != 0);

    for (int n = n0; n < N_; n += nStride) {
        // ---- Per-lane x inputs for A rows (row M = c, channel f0+c) ----
        float x0, x1[3], x2[5];
        x0 = x[((size_t)(0 * N_ + n) * MI_ + 0) * F_ + fA];
#pragma unroll
        for (int m = 0; m < 3; ++m) x1[m] = x[((size_t)(1 * N_ + n) * MI_ + m) * F_ + fA];
#pragma unroll
        for (int m = 0; m < 5; ++m) x2[m] = x[((size_t)(2 * N_ + n) * MI_ + m) * F_ + fA];

        // ---- A-row products, K = 0..31 (s / k folded into A) ----
        float prod[32];
        const float sxA  = s01 * x0;
        const float sxB  = s02 * x0;
        const float sxC0 = s12 * x1[0];
        const float sxC1 = s12 * x1[1];
        const float sxC2 = s12 * x1[2];
        prod[0] = sxA * x1[0]; prod[1] = sxA * x1[1]; prod[2] = sxA * x1[2];
#pragma unroll
        for (int q = 0; q < 5; ++q) prod[3 + q]  = sxB  * x2[q];
#pragma unroll
        for (int q = 0; q < 5; ++q) prod[8 + q]  = sxC0 * x2[q];
#pragma unroll
        for (int q = 0; q < 5; ++q) prod[13 + q] = sxC1 * x2[q];
#pragma unroll
        for (int q = 0; q < 5; ++q) prod[18 + q] = sxC2 * x2[q];
        prod[23] = x0    * kk0;
        prod[24] = x1[0] * kk1; prod[25] = x1[1] * kk1; prod[26] = x1[2] * kk1;
#pragma unroll
        for (int m = 0; m < 5; ++m) prod[27 + m] = x2[m] * kk2;

        // ---- 8 fp32 WMMA steps in two independent 4-deep chains ----
        v8f acc0 = {};
        v8f acc1 = {};
#pragma unroll
        for (int t = 0; t < 4; ++t) {
            v2f A;
            A.x = h ? prod[4 * t + 2] : prod[4 * t + 0];
            A.y = h ? prod[4 * t + 3] : prod[4 * t + 1];
            acc0 = __builtin_amdgcn_wmma_f32_16x16x4_f32(false, A, false, B[t],
                                                         (short)0, acc0, false, false);
        }
#pragma unroll
        for (int t = 4; t < 8; ++t) {
            v2f A;
            A.x = h ? prod[4 * t + 2] : prod[4 * t + 0];
            A.y = h ? prod[4 * t + 3] : prod[4 * t + 1];
            acc1 = __builtin_amdgcn_wmma_f32_16x16x4_f32(false, A, false, B[t],
                                                         (short)0, acc1, false, false);
        }
        const v8f acc = acc0 + acc1;

        // ---- Store D: lane holds column c, rows M = v + 8*hi -> f = f0 + 8*hi + v ----
        {
            float* base = out + (((size_t)lD * N_ + n) * MO_ + mD) * F_ + f0 + 8 * hi;
            f4 lo  = { acc[0], acc[1], acc[2], acc[3] };
            f4 hi4 = { acc[4], acc[5], acc[6], acc[7] };
            *(f4*)(base + 0) = lo;
            *(f4*)(base + 4) = hi4;
        }

        // ---- Zero the 12 invalid (l,m) output slots for this tile's 16 channels ----
        if (doZero) {
            float* zb = out + (((size_t)lZ * N_ + n) * MO_ + mZ) * F_ + f0 + 8 * hf;
            f4 z = { 0.f, 0.f, 0.f, 0.f };
            *(f4*)(zb + 0) = z;
            *(f4*)(zb + 4) = z;
        }
    }
}

extern "C" void kernel_launch(void* const* d_in, const int* in_sizes, int n_in,
                              void* d_out, int out_size, void* d_ws, size_t ws_size,
                              hipStream_t stream) {
    const float* x   = (const float*)d_in[0];
    const float* k   = (const float*)d_in[1];
    const float* s   = (const float*)d_in[2];
    const float* cgc = (const float*)d_in[3];
    float* out = (float*)d_out;

    // 32768 waves (4096 blocks x 8 waves); each wave keeps a fixed 16-channel slice
    // (so B/s/k are loop-invariant) and iterates over 8 points.
    dim3 block(256);
    dim3 grid(4096);
    hipLaunchKernelGGL(selfmix_wmma_kernel, grid, block, 0, stream, x, k, s, cgc, out);
}